// CenternetPostprocess_54666343743493
// MI455X (gfx1250) — compile-verified
//
#include <hip/hip_runtime.h>
#include <hip/hip_bf16.h>
#include <math.h>

#ifndef __has_builtin
#define __has_builtin(x) 0
#endif

#define AS1 __attribute__((address_space(1)))
#define AS3 __attribute__((address_space(3)))

typedef int v4i __attribute__((ext_vector_type(4)));

constexpr int NCLS   = 80;
constexpr int NCH4   = 84;          // 80 classes + 4 coors
constexpr int W      = 128;
constexpr int H      = 128;
constexpr int HW     = W * H;
constexpr int BATCH  = 32;
constexpr int KDET   = 100;
constexpr int TPB    = 256;
constexpr int CHUNK  = W * NCLS;    // 10240 = one row y, all classes, transposed order
constexpr int SLOTS1 = CHUNK / TPB; // 40
constexpr int NCHUNK = H;           // 128 row-chunks per batch
constexpr int CAND   = NCHUNK * KDET; // 12800 candidates per batch
constexpr int SLOTS2 = CAND / TPB;  // 50

// wave32 argmax on (score, index) with jax tie-break (smaller index wins on equal score)
__device__ __forceinline__ void wave_argmax(float &s, int &j) {
#pragma unroll
  for (int m = 16; m > 0; m >>= 1) {
    float s2 = __shfl_xor(s, m, 32);
    int   j2 = __shfl_xor(j, m, 32);
    if (s2 > s || (s2 == s && j2 < j)) { s = s2; j = j2; }
  }
}

// ---------------------------------------------------------------------------
// Stage 1: one block per (row y, batch b). Computes 3x3-NMS scores for the
// 10240 transposed-flat elements of that row, keeps them in registers, and
// emits the row-local top-100 (score, global transposed index) candidates.
//
// Branchless NMS: clamp out-of-bounds neighbor coords to the edge. Clamped
// coords duplicate in-window elements, and the reference pads with -inf, so
// max(9 clamped loads) == hmax exactly; keep <=> (m9 <= v).
// ---------------------------------------------------------------------------
__global__ __launch_bounds__(TPB) void nms_rowtopk(const float* __restrict__ yp,
                                                   float* __restrict__ candS,
                                                   int*   __restrict__ candI) {
  const int yrow = blockIdx.x;
  const int b    = blockIdx.y;
  const int t    = threadIdx.x;
  const int x    = t & (W - 1);   // fixed per thread
  const int chi  = t >> 7;        // class parity offset (0/1)
  const int basej = x * NCLS + chi;   // j_k = basej + 2k  (monotone in k)

  const int xm = (x > 0)     ? x - 1 : x;        // per-thread clamp (one select)
  const int xp = (x < W - 1) ? x + 1 : x;
  const int y0 = (yrow > 0)     ? yrow - 1 : yrow;  // uniform per block
  const int y2 = (yrow < H - 1) ? yrow + 1 : yrow;

  const float* bp = yp + (size_t)b * NCH4 * HW;

  float loc[SLOTS1];
  float ms = -INFINITY; int mk = 0;

#pragma unroll
  for (int k = 0; k < SLOTS1; ++k) {
    const int c = 2 * k + chi;
    const float* pl = bp + (size_t)c * HW;
    const float* r0 = pl + y0 * W;
    const float* r1 = pl + yrow * W;
    const float* r2 = pl + y2 * W;
    const float v  = r1[x];
    // 9 unconditional coalesced loads; m9 == hmax (clamped dups are in-window)
    const float m9 = fmaxf(
        fmaxf(fmaxf(r0[xm], r0[x]), fmaxf(r0[xp], r1[xm])),
        fmaxf(fmaxf(v, r1[xp]),
              fmaxf(fmaxf(r2[xm], r2[x]), r2[xp])));
    // keep iff this element equals the 3x3 max; suppressed -> v*0 (matches hm*keep incl. -0)
    const float s = (m9 <= v) ? v : v * 0.0f;
    loc[k] = s;
    if (s > ms) { ms = s; mk = k; }   // strict > keeps smallest k on ties
  }

  __shared__ float wm[TPB / 32];
  __shared__ int   wj[TPB / 32];
  const int wid = t >> 5, lane = t & 31;
  const int cbase = (b * NCHUNK + yrow) * KDET;

  for (int it = 0; it < KDET; ++it) {
    float s = ms;
    int   j = basej + 2 * mk;
    wave_argmax(s, j);
    if (lane == 0) { wm[wid] = s; wj[wid] = j; }
    __syncthreads();
    float bs = wm[0]; int bj = wj[0];
#pragma unroll
    for (int w = 1; w < TPB / 32; ++w) {
      const float s2 = wm[w]; const int j2 = wj[w];
      if (s2 > bs || (s2 == bs && j2 < bj)) { bs = s2; bj = j2; }
    }
    if (t == 0) {
      candS[cbase + it] = bs;
      candI[cbase + it] = yrow * CHUNK + bj;  // global transposed flat index in batch
    }
    // owner thread removes winner and rescans its registers
    const int bx = bj / NCLS;
    const int bc = bj - bx * NCLS;
    const int owner = ((bc & 1) << 7) | bx;
    if (t == owner) {
      const int kw = bc >> 1;
      ms = -INFINITY; mk = 0;
#pragma unroll
      for (int k = 0; k < SLOTS1; ++k) {
        const float v = (k == kw) ? -INFINITY : loc[k];
        loc[k] = v;
        if (v > ms) { ms = v; mk = k; }
      }
    }
    __syncthreads();
  }
}

// ---------------------------------------------------------------------------
// Stage 2: one block per batch. Async-copies the 12800 candidate scores into
// LDS (gfx1250 GLOBAL_LOAD_ASYNC_TO_LDS path), selects the final top-100 with
// the same register-resident argmax, then 100 threads do the coors gather and
// box math in parallel.
// ---------------------------------------------------------------------------
__global__ __launch_bounds__(TPB) void final_topk(const float* __restrict__ yp,
                                                  const float* __restrict__ candS,
                                                  const int*   __restrict__ candI,
                                                  float* __restrict__ out) {
  const int b = blockIdx.x;
  const int t = threadIdx.x;

  __shared__ float sc[CAND];        // 51.2 KB (320 KB/WGP on CDNA5)
  __shared__ float wm[TPB / 32];
  __shared__ int   wj[TPB / 32];
  __shared__ float winS[KDET];
  __shared__ int   winP[KDET];

  const float* cs = candS + (size_t)b * CAND;

#if __has_builtin(__builtin_amdgcn_global_load_async_to_lds_b128)
  for (int i = t; i < CAND / 4; i += TPB) {
    __builtin_amdgcn_global_load_async_to_lds_b128(
        (AS1 v4i*)(cs + 4 * i), (AS3 v4i*)(&sc[4 * i]), 0, 0);
  }
#if __has_builtin(__builtin_amdgcn_s_wait_asynccnt)
  __builtin_amdgcn_s_wait_asynccnt(0);
#else
  asm volatile("s_wait_asynccnt 0" ::: "memory");
#endif
#else
  for (int i = t; i < CAND; i += TPB) sc[i] = cs[i];
#endif
  __syncthreads();

  float loc[SLOTS2];
  float ms = -INFINITY; int mk = 0;
  const int pbase = t * SLOTS2;
#pragma unroll
  for (int k = 0; k < SLOTS2; ++k) {
    const float s = sc[pbase + k];
    loc[k] = s;
    if (s > ms) { ms = s; mk = k; }
  }

  const int wid = t >> 5, lane = t & 31;
  for (int it = 0; it < KDET; ++it) {
    float s = ms;
    int   p = pbase + mk;
    wave_argmax(s, p);
    if (lane == 0) { wm[wid] = s; wj[wid] = p; }
    __syncthreads();
    float bs = wm[0]; int bp = wj[0];
#pragma unroll
    for (int w = 1; w < TPB / 32; ++w) {
      const float s2 = wm[w]; const int p2 = wj[w];
      if (s2 > bs || (s2 == bs && p2 < bp)) { bs = s2; bp = p2; }
    }
    if (t == 0) { winS[it] = bs; winP[it] = bp; }
    const int owner = bp / SLOTS2;
    if (t == owner) {
      const int kw = bp - pbase;
      ms = -INFINITY; mk = 0;
#pragma unroll
      for (int k = 0; k < SLOTS2; ++k) {
        const float v = (k == kw) ? -INFINITY : loc[k];
        loc[k] = v;
        if (v > ms) { ms = v; mk = k; }
      }
    }
    __syncthreads();
  }

  // Parallel epilogue: detection t gathers coors and writes the 8 outputs.
  if (t < KDET) {
    const int   wp = winP[t];
    const float sc0 = winS[t];
    const int gi = candI[(size_t)b * CAND + wp];
    const int p  = gi / NCLS;
    const int c  = gi - p * NCLS;
    const float xs = (float)(p & (W - 1));
    const float ys = (float)(p >> 7);
    const float* cp = yp + ((size_t)b * NCH4 + NCLS) * HW + p;
    const float g0 = cp[0], g1 = cp[HW], g2 = cp[2 * HW], g3 = cp[3 * HW];
    float* o = out + (size_t)(b * KDET + t) * 8;
    o[0] = (float)c + 1.0f;
    o[1] = sc0;
    o[2] = (4.0f * xs - g0) * (1.0f / 512.0f);
    o[3] = (4.0f * ys - g1) * (1.0f / 512.0f);
    o[4] = (4.0f * xs + g2) * (1.0f / 512.0f);
    o[5] = (4.0f * ys + g3) * (1.0f / 512.0f);
    o[6] = ys;
    o[7] = xs;
  }
}

extern "C" void kernel_launch(void* const* d_in, const int* in_sizes, int n_in,
                              void* d_out, int out_size, void* d_ws, size_t ws_size,
                              hipStream_t stream) {
  const float* yp = (const float*)d_in[0];
  float* out = (float*)d_out;
  float* candS = (float*)d_ws;                                   // 32*12800 f32 = 1.64 MB
  int*   candI = (int*)((char*)d_ws + (size_t)BATCH * CAND * 4); // 32*12800 i32 = 1.64 MB

  dim3 g1(NCHUNK, BATCH);
  hipLaunchKernelGGL(nms_rowtopk, g1, dim3(TPB), 0, stream, yp, candS, candI);
  hipLaunchKernelGGL(final_topk, dim3(BATCH), dim3(TPB), 0, stream, yp, candS, candI, out);
}